// Group_14482629722271
// MI455X (gfx1250) — compile-verified
//
#include <hip/hip_runtime.h>
#include <hip/hip_bf16.h>

typedef __attribute__((ext_vector_type(2))) float v2f;
typedef __attribute__((ext_vector_type(4))) float v4f;
typedef __attribute__((ext_vector_type(8))) float v8f;

#define BATCH 4
#define N1    2048
#define N2    8192
#define NCH   64      // feature channels
#define NOUTC 67      // 3 + NCH
#define KNN   32
#define TM    16      // queries per block (one WMMA M tile)
#define CHUNK 512     // source points per LDS distance tile

// One wave (32 lanes) per block: EXEC all-1s so WMMA is legal everywhere.
__global__ __launch_bounds__(32) void knn_group_wmma_kernel(
    const float* __restrict__ xyz2,   // [B,3,N2]
    const float* __restrict__ xyz1,   // [B,3,N1]
    const float* __restrict__ feats,  // [B,NCH,N2]
    float* __restrict__ out)          // [B,67,N1,KNN]
{
    // distance tile stored transposed: [local_n][16 queries] so each lane's
    // 8 WMMA results are contiguous -> 2x ds_store_b128 per tile.
    __shared__ float distLDS[CHUNK * TM];   // 32 KB
    __shared__ float topd[TM][KNN];         // running k-smallest distances
    __shared__ int   topi[TM][KNN];         // and their indices
    __shared__ float yyLDS[TM];             // |query|^2 per query

    const int lane = threadIdx.x;                 // 0..31
    const int blk  = blockIdx.x;                  // b * (N1/TM) + mtile
    const int b    = blk / (N1 / TM);
    const int m0   = (blk % (N1 / TM)) * TM;

    const float* x1 = xyz1 + (size_t)b * 3 * N1;
    const float* x2 = xyz2 + (size_t)b * 3 * N2;

    // ---- A matrix: -2 * query coords, 16x4 f32, K=3 zero-padded ----
    // ISA layout (32-bit A 16x4): lanes 0-15 M=lane hold K=0 (v0), K=1 (v1);
    // lanes 16-31 M=lane-16 hold K=2 (v0), K=3 (v1).
    const int am = lane & 15;
    const float q0 = x1[0 * N1 + m0 + am];
    const float q1 = x1[1 * N1 + m0 + am];
    const float q2 = x1[2 * N1 + m0 + am];
    v2f Amat;
    if (lane < 16) { Amat.x = -2.0f * q0; Amat.y = -2.0f * q1; }
    else           { Amat.x = -2.0f * q2; Amat.y = 0.0f;       }

    if (lane < 16) {
        yyLDS[lane] = q0 * q0 + q1 * q1 + q2 * q2;
        for (int j = 0; j < KNN; ++j) { topd[lane][j] = 3.4e38f; topi[lane][j] = 0; }
    }
    __syncthreads();

    // Hoist per-lane C-matrix row bases (|query|^2 for this lane's 8 rows)
    // out of the tile loop: 8 LDS reads once per block instead of per tile.
    const int mbase = (lane < 16) ? 0 : 8;
    float cbase[8];
    #pragma unroll
    for (int v = 0; v < 8; ++v) cbase[v] = yyLDS[mbase + v];

    // per-lane worst-entry tracker (lanes 0-15 only; each owns query m=lane)
    float worstVal = 3.4e38f;
    int   worstPos = 0;

    for (int base = 0; base < N2; base += CHUNK) {
        // ---- compute 16 x CHUNK distance tile via WMMA f32 16x16x4 ----
        for (int t = 0; t < CHUNK; t += 16) {
            const int nl = lane & 15;           // N = lane % 16
            const int n  = base + t + nl;
            const float p0 = x2[0 * N2 + n];
            const float p1 = x2[1 * N2 + n];
            const float p2 = x2[2 * N2 + n];
            const float xx = p0 * p0 + p1 * p1 + p2 * p2;

            // B matrix 4x16 f32: lanes 0-15 hold rows K=0 (v0), K=1 (v1);
            // lanes 16-31 hold rows K=2 (v0), K=3 (v1).
            v2f Bmat;
            if (lane < 16) { Bmat.x = p0; Bmat.y = p1; }
            else           { Bmat.x = p2; Bmat.y = 0.0f; }

            // C[m,n] = |query_m|^2 + |src_n|^2 ; D = A*B + C = sq-dist tile
            v8f Cmat;
            #pragma unroll
            for (int v = 0; v < 8; ++v) Cmat[v] = cbase[v] + xx;

            v8f D = __builtin_amdgcn_wmma_f32_16x16x4_f32(
                false, Amat, false, Bmat, (short)0, Cmat, false, false);

            // maximumNumber(d, 0): one v_max_num_f32 does NaN->0 AND clamp>=0
            v4f lo, hi;
            #pragma unroll
            for (int v = 0; v < 4; ++v) {
                lo[v] = fmaxf(D[v],     0.0f);
                hi[v] = fmaxf(D[v + 4], 0.0f);
            }
            // contiguous 8 floats per lane -> 2x ds_store_b128
            *(v4f*)&distLDS[(t + nl) * TM + mbase]     = lo;
            *(v4f*)&distLDS[(t + nl) * TM + mbase + 4] = hi;
        }
        __syncthreads();

        // ---- incremental top-32 (smallest): lanes 0-15, one query each ----
        if (lane < 16) {
            for (int j = 0; j < CHUNK; ++j) {
                const float d = distLDS[j * TM + lane];  // 16 lanes -> 16 consecutive DWORDs
                if (d < worstVal) {
                    topd[lane][worstPos] = d;
                    topi[lane][worstPos] = base + j;
                    float w = -1.0f; int wp = 0;
                    for (int t2 = 0; t2 < KNN; ++t2) {
                        const float v = topd[lane][t2];
                        if (v > w) { w = v; wp = t2; }
                    }
                    worstVal = w; worstPos = wp;
                }
            }
        }
        __syncthreads();
    }

    // ---- sort each query's 32 entries ascending (matches top_k order) ----
    if (lane < 16) {
        for (int i = 1; i < KNN; ++i) {
            const float dv = topd[lane][i]; const int iv = topi[lane][i];
            int j = i - 1;
            while (j >= 0 && topd[lane][j] > dv) {
                topd[lane][j + 1] = topd[lane][j];
                topi[lane][j + 1] = topi[lane][j];
                --j;
            }
            topd[lane][j + 1] = dv; topi[lane][j + 1] = iv;
        }
    }
    __syncthreads();

    // ---- gather + write output: out[b][c][n1][k], lane = neighbor j ----
    const float* f  = feats + (size_t)b * NCH * N2;
    float*       ob = out   + (size_t)b * NOUTC * N1 * KNN;
    for (int m = 0; m < TM; ++m) {
        const int n1  = m0 + m;
        const int idx = topi[m][lane];
        #pragma unroll
        for (int c = 0; c < 3; ++c) {
            const float val = x2[c * N2 + idx] - x1[c * N1 + n1];
            ob[((size_t)c * N1 + n1) * KNN + lane] = val;   // coalesced over lane
        }
        for (int c = 0; c < NCH; ++c) {
            ob[((size_t)(3 + c) * N1 + n1) * KNN + lane] = f[c * N2 + idx];
        }
    }
}

extern "C" void kernel_launch(void* const* d_in, const int* in_sizes, int n_in,
                              void* d_out, int out_size, void* d_ws, size_t ws_size,
                              hipStream_t stream) {
    const float* xyz2  = (const float*)d_in[0];  // [4,3,8192]
    const float* xyz1  = (const float*)d_in[1];  // [4,3,2048]
    const float* feats = (const float*)d_in[2];  // [4,64,8192]
    float* out = (float*)d_out;                  // [4,67,2048,32]

    const int blocks = BATCH * (N1 / TM);        // 512 blocks, 1 wave each
    knn_group_wmma_kernel<<<blocks, 32, 0, stream>>>(xyz2, xyz1, feats, out);
}